// ImprovedGNN_80977313398919
// MI455X (gfx1250) — compile-verified
//
#include <hip/hip_runtime.h>
#include <math.h>

typedef __attribute__((ext_vector_type(2))) float v2f;
typedef __attribute__((ext_vector_type(8))) float v8f;

// ---------------------------------------------------------------------------
// WMMA f32 GEMM:  out[M,NOUT] = A[M,K] @ W[K,NOUT] (+ bias[NOUT] if non-null)
// One wave -> one 16x16 output tile, K stepped by 4 via V_WMMA_F32_16X16X4_F32.
// The 16 output columns' weight slab W[:, n0:n0+16] is staged in LDS and
// shared by all 8 waves of the block.
// ---------------------------------------------------------------------------
template <int K, int NOUT>
__global__ __launch_bounds__(256) void gemm_wmma_f32(
    const float* __restrict__ A, const float* __restrict__ W,
    const float* __restrict__ bias, float* __restrict__ out, int M) {
  __shared__ float Ws[K * 16];
  const int n0 = blockIdx.y * 16;
  for (int i = threadIdx.x; i < K * 16; i += 256) {
    const int k = i >> 4, nn = i & 15;
    Ws[i] = W[k * NOUT + n0 + nn];
  }
  __syncthreads();

  const int wave = threadIdx.x >> 5;
  const int lane = threadIdx.x & 31;
  const int tile_m = blockIdx.x * 8 + wave;
  if (tile_m * 16 >= M) return;

  const int r  = lane & 15;          // A-row / B-col / C-col within tile
  const int kb = (lane >> 4) << 1;   // lanes 0-15 hold K=k,k+1 ; 16-31 hold K=k+2,k+3
  const float* Arow = A + (size_t)(tile_m * 16 + r) * K;

  v8f c = {};
#pragma unroll 4
  for (int k = 0; k < K; k += 4) {
    v2f a, b;
    a.x = Arow[k + kb];
    a.y = Arow[k + kb + 1];
    b.x = Ws[(k + kb) * 16 + r];
    b.y = Ws[(k + kb + 1) * 16 + r];
    c = __builtin_amdgcn_wmma_f32_16x16x4_f32(false, a, false, b, (short)0, c,
                                              false, false);
  }

  const int col = n0 + r;
  const float bv = bias ? bias[col] : 0.0f;
  const int rbase = tile_m * 16 + ((lane >> 4) << 3);  // +8 rows for upper lanes
#pragma unroll
  for (int i = 0; i < 8; ++i)
    out[(size_t)(rbase + i) * NOUT + col] = c[i] + bv;
}

// ---------------------------------------------------------------------------
// BatchNorm (training-mode over node dim) split into stats / finalize / apply
// ---------------------------------------------------------------------------
__global__ void bn_stats(const float* __restrict__ x, float* __restrict__ sums,
                         float* __restrict__ sumsq, int n, int C) {
  const int gt = blockIdx.x * blockDim.x + threadIdx.x;
  const int c = gt % C;
  int r = gt / C;
  const int rstride = (gridDim.x * blockDim.x) / C;  // divisible: 131072 % C == 0
  float s = 0.f, sq = 0.f;
  for (; r < n; r += rstride) {
    const float v = x[(size_t)r * C + c];
    s += v;
    sq += v * v;
  }
  atomicAdd(&sums[c], s);
  atomicAdd(&sumsq[c], sq);
}

__global__ void bn_finalize(const float* sums, const float* sumsq,
                            const float* __restrict__ gamma,
                            const float* __restrict__ beta, float* scale,
                            float* shift, int n, int C) {
  const int c = blockIdx.x * blockDim.x + threadIdx.x;
  if (c >= C) return;
  const float inv_n = 1.0f / (float)n;
  const float mu = sums[c] * inv_n;
  const float var = sumsq[c] * inv_n - mu * mu;
  const float sc = gamma[c] * rsqrtf(var + 1e-5f);
  scale[c] = sc;
  shift[c] = beta[c] - mu * sc;
}

__global__ void bn_relu_apply(float* __restrict__ x, const float* scale,
                              const float* shift, long total, int C) {
  const long t = (long)blockIdx.x * blockDim.x + threadIdx.x;
  if (t >= total) return;
  const int c = (int)(t % C);
  const float v = x[t] * scale[c] + shift[c];
  x[t] = v > 0.f ? v : 0.f;
}

// ---------------------------------------------------------------------------
// Small utility fills
// ---------------------------------------------------------------------------
__global__ void fill_value(float* __restrict__ p, float v, long count) {
  const long t = (long)blockIdx.x * blockDim.x + threadIdx.x;
  if (t < count) p[t] = v;
}

__global__ void fill_bias_rows(float* __restrict__ out,
                               const float* __restrict__ bias, long total,
                               int C) {
  const long t = (long)blockIdx.x * blockDim.x + threadIdx.x;
  if (t < total) out[t] = bias[t % C];
}

// ---------------------------------------------------------------------------
// GAT attention pipeline
// ---------------------------------------------------------------------------
__global__ void att_scores(const float* __restrict__ hfeat,
                           const float* __restrict__ att,
                           float* __restrict__ a_out, int n, int heads, int D) {
  const int t = blockIdx.x * blockDim.x + threadIdx.x;
  if (t >= n * heads) return;
  const int hd = t % heads;
  const int node = t / heads;
  const float* hp = hfeat + ((size_t)node * heads + hd) * D;
  const float* ap = att + (size_t)hd * D;
  float s = 0.f;
  for (int d = 0; d < D; d += 4)
    s += hp[d] * ap[d] + hp[d + 1] * ap[d + 1] + hp[d + 2] * ap[d + 2] +
         hp[d + 3] * ap[d + 3];
  a_out[t] = s;  // layout [node, head]
}

__device__ __forceinline__ void edge_endpoints(const int* __restrict__ ei,
                                               int idx, int E, int& s, int& d) {
  if (idx < E) {
    s = ei[idx];
    d = ei[E + idx];
  } else {  // self loop
    s = d = idx - E;
  }
}

__device__ __forceinline__ void atomic_max_f32(float* addr, float v) {
  if (v >= 0.f)
    atomicMax((int*)addr, __float_as_int(v));
  else
    atomicMin((unsigned int*)addr, __float_as_uint(v));
}

__global__ void edge_max(const int* __restrict__ ei,
                         const float* __restrict__ a_src,
                         const float* __restrict__ a_dst,
                         float* __restrict__ emax, int E, int etot, int heads) {
  const int t = blockIdx.x * blockDim.x + threadIdx.x;
  if (t >= etot * heads) return;
  const int hd = t % heads;
  const int idx = t / heads;
  int s, d;
  edge_endpoints(ei, idx, E, s, d);
  float e = a_src[s * heads + hd] + a_dst[d * heads + hd];
  e = e > 0.f ? e : 0.2f * e;  // leaky_relu(0.2)
  atomic_max_f32(&emax[d * heads + hd], e);
}

__global__ void edge_expsum(const int* __restrict__ ei,
                            const float* __restrict__ a_src,
                            const float* __restrict__ a_dst,
                            const float* __restrict__ emax,
                            float* __restrict__ ebuf,
                            float* __restrict__ denom, int E, int etot,
                            int heads) {
  const int t = blockIdx.x * blockDim.x + threadIdx.x;
  if (t >= etot * heads) return;
  const int hd = t % heads;
  const int idx = t / heads;
  int s, d;
  edge_endpoints(ei, idx, E, s, d);
  float e = a_src[s * heads + hd] + a_dst[d * heads + hd];
  e = e > 0.f ? e : 0.2f * e;
  const float ex = __expf(e - emax[d * heads + hd]);
  ebuf[t] = ex;
  atomicAdd(&denom[d * heads + hd], ex);
}

__global__ __launch_bounds__(256) void edge_aggregate(
    const int* __restrict__ ei, const float* __restrict__ hfeat,
    const float* __restrict__ ebuf, const float* __restrict__ denom,
    float* __restrict__ out, int E, int etot, int heads, int D) {
  const long t = (long)blockIdx.x * blockDim.x + threadIdx.x;
  const int chunks = D >> 2;
  if (t >= (long)etot * heads * chunks) return;
  const int ch = (int)(t % chunks);
  const long tmp = t / chunks;
  const int hd = (int)(tmp % heads);
  const int idx = (int)(tmp / heads);
  int s, d;
  edge_endpoints(ei, idx, E, s, d);
  const float alpha =
      ebuf[(size_t)idx * heads + hd] / (denom[d * heads + hd] + 1e-16f);
  const float4 hv =
      *(const float4*)(hfeat + ((size_t)s * heads + hd) * D + ch * 4);
  float* op = out + ((size_t)d * heads + hd) * D + ch * 4;
  atomicAdd(op + 0, hv.x * alpha);
  atomicAdd(op + 1, hv.y * alpha);
  atomicAdd(op + 2, hv.z * alpha);
  atomicAdd(op + 3, hv.w * alpha);
}

// ---------------------------------------------------------------------------
// Final projection: out[n] = h[n,:D] . w[:D] + b
// ---------------------------------------------------------------------------
__global__ void final_proj(const float* __restrict__ h,
                           const float* __restrict__ w,
                           const float* __restrict__ b, float* __restrict__ out,
                           int n, int D) {
  const int t = blockIdx.x * blockDim.x + threadIdx.x;
  if (t >= n) return;
  const float* hp = h + (size_t)t * D;
  float s = b[0];
  for (int d = 0; d < D; d += 4)
    s += hp[d] * w[d] + hp[d + 1] * w[d + 1] + hp[d + 2] * w[d + 2] +
         hp[d + 3] * w[d + 3];
  out[t] = s;
}

// ---------------------------------------------------------------------------
static inline long cdl(long a, long b) { return (a + b - 1) / b; }

extern "C" void kernel_launch(void* const* d_in, const int* in_sizes, int n_in,
                              void* d_out, int out_size, void* d_ws,
                              size_t ws_size, hipStream_t stream) {
  const float* x    = (const float*)d_in[0];
  const int* ei     = (const int*)d_in[1];
  const float* W_in = (const float*)d_in[2];
  const float* b_in = (const float*)d_in[3];
  const float* g0   = (const float*)d_in[4];
  const float* be0  = (const float*)d_in[5];
  const float* W1   = (const float*)d_in[6];
  const float* as1  = (const float*)d_in[7];
  const float* ad1  = (const float*)d_in[8];
  const float* bias1= (const float*)d_in[9];
  const float* g1   = (const float*)d_in[10];
  const float* be1  = (const float*)d_in[11];
  const float* W2   = (const float*)d_in[12];
  const float* as2  = (const float*)d_in[13];
  const float* ad2  = (const float*)d_in[14];
  const float* bias2= (const float*)d_in[15];
  const float* g2   = (const float*)d_in[16];
  const float* be2  = (const float*)d_in[17];
  const float* Wf1  = (const float*)d_in[18];
  const float* bf1  = (const float*)d_in[19];
  const float* gf   = (const float*)d_in[20];
  const float* bef  = (const float*)d_in[21];
  const float* Wf2  = (const float*)d_in[22];
  const float* bf2  = (const float*)d_in[23];

  const int N = in_sizes[0] / 16;   // 50000 (== 3125 * 16, no edge tiles)
  const int E = in_sizes[1] / 2;    // 800000
  const int ETOT = E + N;           // + self loops
  const int Hdim = 128;

  // ---- workspace layout (floats) ----
  float* wsf   = (float*)d_ws;
  float* bufA  = wsf;                         // N*256, ping
  float* bufB  = bufA + (size_t)N * 256;      // N*256, pong
  float* a_src = bufB + (size_t)N * 256;      // N*2
  float* a_dst = a_src + (size_t)N * 2;       // N*2
  float* emax  = a_dst + (size_t)N * 2;       // N*2
  float* denom = emax + (size_t)N * 2;        // N*2
  float* ebuf  = denom + (size_t)N * 2;       // ETOT*2
  float* sums  = ebuf + (size_t)ETOT * 2;     // 256
  float* sumsq = sums + 256;                  // 256
  float* scale = sumsq + 256;                 // 256
  float* shift = scale + 256;                 // 256

  const dim3 blk(256);
  const int gx_gemm = (int)cdl(N / 16, 8);

  auto run_bn = [&](float* xb, const float* g, const float* be, int C) {
    fill_value<<<2, blk, 0, stream>>>(sums, 0.f, 512);  // sums+sumsq contiguous
    bn_stats<<<512, blk, 0, stream>>>(xb, sums, sumsq, N, C);
    bn_finalize<<<1, blk, 0, stream>>>(sums, sumsq, g, be, scale, shift, N, C);
    const long tot = (long)N * C;
    bn_relu_apply<<<(int)cdl(tot, 256), blk, 0, stream>>>(xb, scale, shift, tot, C);
  };

  auto run_gat = [&](const float* hfeat, const float* attS, const float* attD,
                     const float* bias, float* out, int heads) {
    const int D = Hdim;
    att_scores<<<(int)cdl((long)N * heads, 256), blk, 0, stream>>>(hfeat, attS, a_src, N, heads, D);
    att_scores<<<(int)cdl((long)N * heads, 256), blk, 0, stream>>>(hfeat, attD, a_dst, N, heads, D);
    fill_value<<<(int)cdl((long)N * heads, 256), blk, 0, stream>>>(emax, -INFINITY, (long)N * heads);
    fill_value<<<(int)cdl((long)N * heads, 256), blk, 0, stream>>>(denom, 0.f, (long)N * heads);
    fill_bias_rows<<<(int)cdl((long)N * heads * D, 256), blk, 0, stream>>>(out, bias, (long)N * heads * D, heads * D);
    edge_max<<<(int)cdl((long)ETOT * heads, 256), blk, 0, stream>>>(ei, a_src, a_dst, emax, E, ETOT, heads);
    edge_expsum<<<(int)cdl((long)ETOT * heads, 256), blk, 0, stream>>>(ei, a_src, a_dst, emax, ebuf, denom, E, ETOT, heads);
    const long tot = (long)ETOT * heads * (D / 4);
    edge_aggregate<<<(int)cdl(tot, 256), blk, 0, stream>>>(ei, hfeat, ebuf, denom, out, E, ETOT, heads, D);
  };

  // ---- layer 0: h0 = relu(bn(x @ W_in + b_in)) -> bufA[:, :128]
  gemm_wmma_f32<16, 128><<<dim3(gx_gemm, 8), blk, 0, stream>>>(x, W_in, b_in, bufA, N);
  run_bn(bufA, g0, be0, 128);

  // ---- GAT layer 1 (heads=2): hfeat1 = h0 @ W1 -> bufB ; aggregate -> bufA
  gemm_wmma_f32<128, 256><<<dim3(gx_gemm, 16), blk, 0, stream>>>(bufA, W1, nullptr, bufB, N);
  run_gat(bufB, as1, ad1, bias1, bufA, 2);
  run_bn(bufA, g1, be1, 256);  // bufA = h1 [N,256]

  // ---- GAT layer 2 (heads=1): hfeat2 = h1 @ W2 -> bufB ; aggregate -> bufA
  gemm_wmma_f32<256, 128><<<dim3(gx_gemm, 8), blk, 0, stream>>>(bufA, W2, nullptr, bufB, N);
  run_gat(bufB, as2, ad2, bias2, bufA, 1);
  run_bn(bufA, g2, be2, 128);  // bufA = h2 [N,128]

  // ---- FFN: h3 = relu(bn(h2 @ Wf1 + bf1)) -> bufB[:, :64]
  gemm_wmma_f32<128, 64><<<dim3(gx_gemm, 4), blk, 0, stream>>>(bufA, Wf1, bf1, bufB, N);
  run_bn(bufB, gf, bef, 64);

  // ---- output: out[n] = h3[n] . Wf2 + bf2
  final_proj<<<(int)cdl(N, 256), blk, 0, stream>>>(bufB, Wf2, bf2, (float*)d_out, N, 64);
}